// Attention_simple_35115652612128
// MI455X (gfx1250) — compile-verified
//
#include <hip/hip_runtime.h>
#include <math.h>

#define DD   512     // context_state_dim
#define CH   16      // tokens per staged chunk
#define BLK  512     // threads per block (16 wave32)

typedef __attribute__((ext_vector_type(2))) float v2f;
typedef __attribute__((ext_vector_type(8))) float v8f;

// ---------------------------------------------------------------------------
// Exclusive scan of segment lengths -> starts (B <= 1024, single block).
// ---------------------------------------------------------------------------
__global__ void seg_starts_kernel(const int* __restrict__ len,
                                  int* __restrict__ starts, int n) {
  __shared__ int sh[1024];
  int t = threadIdx.x;
  int v = (t < n) ? len[t] : 0;
  sh[t] = v;
  __syncthreads();
  for (int off = 1; off < 1024; off <<= 1) {
    int a = (t >= off) ? sh[t - off] : 0;
    __syncthreads();
    sh[t] += a;
    __syncthreads();
  }
  if (t < n) starts[t] = sh[t] - v;   // exclusive prefix
}

// ---------------------------------------------------------------------------
// Issue one CH x DD fp32 chunk (32 KB) as async copies global -> LDS.
// 512 threads x 4 x b128 = 2048 x 16B. Addresses clamped so every wave issues
// exactly 4 instructions (uniform ASYNCcnt), never reading past the tensor.
// ---------------------------------------------------------------------------
__device__ __forceinline__ void async_issue_chunk(const char* cbase,
                                                  size_t rowByte, size_t limit,
                                                  unsigned ldsBase, int tid) {
#pragma unroll
  for (int k = 0; k < 4; ++k) {
    size_t gb = rowByte + (size_t)(tid + k * BLK) * 16u;
    if (gb > limit) gb = limit;                       // clamp: tail-safe
    unsigned lds = ldsBase + (unsigned)(tid + k * BLK) * 16u;
    const char* gp = cbase + gb;
    asm volatile("global_load_async_to_lds_b128 %0, %1, off"
                 :: "v"(lds), "v"(gp) : "memory");
  }
}

// ---------------------------------------------------------------------------
// Chunk body, specialized on FULL (nr == CH at compile time) so the hot
// path has no per-row exec-mask guards around the NT stores.
// ---------------------------------------------------------------------------
template <bool FULL>
__device__ __forceinline__ void chunk_compute(
    const float* __restrict__ buf, float* __restrict__ out,
    int row0, int nr, const float* th, const v2f* triA,
    float* carry, float& m, float& den,
    float* sArr, float* wA, float* rdA, float* scShp,
    int tid, int wv, int lane, int nIdx, int h) {
  // ---- scores: wave w handles row w (wave32 shuffle reduction) ----
  if (FULL || wv < nr) {
    float p = 0.0f;
#pragma unroll
    for (int k = 0; k < 16; ++k) p += buf[wv * DD + lane + 32 * k] * th[k];
#pragma unroll
    for (int o = 16; o > 0; o >>= 1) p += __shfl_xor(p, o, 32);
    if (lane == 0) sArr[wv] = p;
  }
  __syncthreads();

  // ---- scalar: chunk max, rescale factor, weights, 1/den per token ----
  if (tid == 0) {
    const int lim = FULL ? CH : nr;
    float cm = -3.4e38f;
    for (int t = 0; t < lim; ++t) cm = fmaxf(cm, sArr[t]);
    float mN = fmaxf(m, cm);
    float sc = __expf(m - mN);       // rescale old carry/den into new frame
    m = mN;
    den *= sc;
    for (int t = 0; t < CH; ++t) {
      float w = (FULL || t < nr) ? __expf(sArr[t] - m) : 0.0f;
      wA[t] = w;
      den  += w;
      rdA[t] = 1.0f / den;
    }
    *scShp = sc;
  }
  __syncthreads();

  // ---- WMMA prefix scan: each wave owns two 16-column tiles ----
  const float scv = *scShp;
#pragma unroll
  for (int tile = 0; tile < 2; ++tile) {
    const int col = (wv * 2 + tile) * 16 + nIdx;
    v8f acc;
#pragma unroll
    for (int r = 0; r < 8; ++r) acc[r] = carry[tile] * scv;
#pragma unroll
    for (int s = 0; s < 4; ++s) {
      const int kA = 4 * s + 2 * h;            // K indices j=0,1 of this lane
      v2f bv;
      bv.x = wA[kA + 0] * buf[(kA + 0) * DD + col];
      bv.y = wA[kA + 1] * buf[(kA + 1) * DD + col];
      acc = __builtin_amdgcn_wmma_f32_16x16x4_f32(
          false, triA[s], false, bv, (short)0, acc, false, false);
    }
    // store: row = r + 8*h, col fixed per lane; NT (never re-read)
    float* ob = out + (size_t)row0 * DD + col;
#pragma unroll
    for (int r = 0; r < 8; ++r) {
      const int t = r + 8 * h;
      if (FULL || t < nr)
        __builtin_nontemporal_store(acc[r] * rdA[t], ob + (size_t)t * DD);
    }
    // new carry = row 15 of D (lives in VGPR7 of the upper half-wave)
    carry[tile] = __shfl(acc[7], 16 + nIdx, 32);
  }
}

// ---------------------------------------------------------------------------
// One workgroup per segment; single pass (online softmax).
// Intra-chunk inclusive prefix sums of w[t]*x[t,d] computed as
//   D = TriL(16x16) x W(16x16-tile) + C(carry)   via V_WMMA_F32_16X16X4_F32.
// ---------------------------------------------------------------------------
__global__ __launch_bounds__(BLK) void seg_softmax_scan_kernel(
    const float* __restrict__ ctx, const float* __restrict__ theta,
    const int* __restrict__ lengths, const int* __restrict__ starts,
    float* __restrict__ out, long long totalTok) {
  extern __shared__ float smem[];                 // 2 * CH * DD floats (64 KB)
  __shared__ float sArr[CH], wA[CH], rdA[CH], scSh;

  const int b    = blockIdx.x;
  const int tid  = threadIdx.x;
  const int wv   = tid >> 5;
  const int lane = tid & 31;
  const int nIdx = lane & 15;        // N / M index within half-wave
  const int h    = lane >> 4;        // half-wave select
  const int start = starts[b];
  const int L     = lengths[b];

  // theta in registers, identical across waves (lane-strided layout)
  float th[16];
#pragma unroll
  for (int k = 0; k < 16; ++k) th[k] = theta[lane + 32 * k];

  // Lower-triangular ones matrix in WMMA A layout (16x4 f32 per step):
  // element (M = lane&15, K = 4*s + j + 2*(lane>>4)) = (K <= M)
  v2f triA[4];
#pragma unroll
  for (int s = 0; s < 4; ++s) {
    triA[s].x = ((4 * s + 0 + 2 * h) <= nIdx) ? 1.0f : 0.0f;
    triA[s].y = ((4 * s + 1 + 2 * h) <= nIdx) ? 1.0f : 0.0f;
  }

  const char*  cbase = (const char*)ctx;
  const size_t limit = (size_t)totalTok * (DD * 4) - 16;
  const unsigned dynBase = (unsigned)(size_t)(void*)smem; // LDS byte offset

  const int nch = (L + CH - 1) / CH;
  int pb = 0;

  async_issue_chunk(cbase, (size_t)start * (DD * 4), limit, dynBase, tid);

  float carry[2] = {0.0f, 0.0f};     // per-column numerator carries (2 tiles)
  float m   = -3.4e38f;              // running max   (thread 0 only)
  float den = 0.0f;                  // running denom (thread 0 only)

  for (int c = 0; c < nch; ++c) {
    const int row0 = start + c * CH;
    int nr = L - c * CH; if (nr > CH) nr = CH;
    const bool hasNext = (c + 1) < nch;

    if (hasNext) {
      async_issue_chunk(cbase, (size_t)(row0 + CH) * (DD * 4), limit,
                        dynBase + (unsigned)((pb ^ 1) * CH * DD * 4), tid);
      asm volatile("s_wait_asynccnt 4" ::: "memory");   // chunk c landed
    } else {
      asm volatile("s_wait_asynccnt 0" ::: "memory");
    }
    __syncthreads();

    const float* buf = smem + (size_t)pb * (CH * DD);

    if (nr == CH) {     // hot path: straight-line, no store guards
      chunk_compute<true>(buf, out, row0, CH, th, triA, carry, m, den,
                          sArr, wA, rdA, &scSh, tid, wv, lane, nIdx, h);
    } else {            // tail chunk of the segment only
      chunk_compute<false>(buf, out, row0, nr, th, triA, carry, m, den,
                           sArr, wA, rdA, &scSh, tid, wv, lane, nIdx, h);
    }

    __syncthreads();   // nobody still reads buf[pb] before it is re-targeted
    pb ^= 1;
  }
}

// ---------------------------------------------------------------------------
// inputs: [0] context (T*D f32), [1] theta (D f32), [2] lengths (B i32),
//         [3] seg_ids (T i32, unused). output: T*D f32.
// ---------------------------------------------------------------------------
extern "C" void kernel_launch(void* const* d_in, const int* in_sizes, int n_in,
                              void* d_out, int out_size, void* d_ws,
                              size_t ws_size, hipStream_t stream) {
  (void)n_in; (void)out_size; (void)ws_size;
  const float* ctx     = (const float*)d_in[0];
  const float* theta   = (const float*)d_in[1];
  const int*   lengths = (const int*)d_in[2];

  const long long T = (long long)in_sizes[0] / DD;   // 131072
  const int       B = in_sizes[2];                   // 1024

  int* starts = (int*)d_ws;                          // 4 KB scratch
  seg_starts_kernel<<<1, 1024, 0, stream>>>(lengths, starts, B);

  const size_t shmem = (size_t)2 * CH * DD * sizeof(float);   // 64 KB
  seg_softmax_scan_kernel<<<B, BLK, shmem, stream>>>(
      ctx, theta, lengths, starts, (float*)d_out, T);
}